// HeteroGraphSAGE_54400055771301
// MI455X (gfx1250) — compile-verified
//
#include <hip/hip_runtime.h>

// ---------------------------------------------------------------------------
// HeteroGraphSAGE on MI455X (gfx1250).
// * GEMMs: V_WMMA_F32_16X16X4_F32 (fp32 == reference precision; GEMM FLOPs are
//   negligible vs the ~4.8 GB edge gather/scatter, so no quantization).
// * A-tiles staged to LDS (padded stride 132 -> conflict-free ds_load_b64
//   fragment reads), via gfx1250 async global->LDS copies (ASYNCcnt-tracked).
// * N is a template parameter (128 / 64) so the fully unrolled K-loop uses
//   immediate-offset loads instead of per-step 64-bit address math.
// * Edge segment-max: relu output >= 0 -> IEEE bits monotone -> u32 atomicMax
//   on bit patterns; zero-init also gives DGL zero-in-degree semantics.
// ---------------------------------------------------------------------------

typedef float v2f __attribute__((ext_vector_type(2)));
typedef float v8f __attribute__((ext_vector_type(8)));
typedef int   v4i_vs __attribute__((vector_size(4 * sizeof(int))));  // builtin's pointee type

#if __has_builtin(__builtin_amdgcn_wmma_f32_16x16x4_f32)
#define HAVE_WMMA_F32X4 1
#else
#define HAVE_WMMA_F32X4 0
#endif

#if __has_builtin(__builtin_amdgcn_global_load_async_to_lds_b128)
#define HAVE_ASYNC_LDS 1
#else
#define HAVE_ASYNC_LDS 0
#endif

#define LDS_STRIDE 132   // 128 + 4 pad: bank(m) = (4*m + k) % 64, conflict-free

#if HAVE_ASYNC_LDS
__device__ __forceinline__ void async_copy16(const float* g, float* l) {
    // signature (from clang diagnostic): (AS1 v4i*, AS3 v4i*, i32 offset, i32 cpol)
    __builtin_amdgcn_global_load_async_to_lds_b128(
        (__attribute__((address_space(1))) v4i_vs*)(g),
        (__attribute__((address_space(3))) v4i_vs*)(l), 0, 0);
}
#endif

// out[M,N] = f( A[M,128] @ W[128,N] (+ A2[M,128] @ W2[128,N]) + bias[N] )
// Block: (N/16) waves; one 16x16 C tile per wave; grid.x = M/16.
// EXEC is all-ones in every wave (WMMA requirement); all dims multiples of 16.
template <int N>
__global__ void gemm128_wmma(const float* __restrict__ A,  const float* __restrict__ W,
                             const float* __restrict__ A2, const float* __restrict__ W2,
                             const float* __restrict__ bias, float* __restrict__ out,
                             int relu) {
    __shared__ __align__(16) float ldsA[2][16 * LDS_STRIDE];
    constexpr int NT = (N / 16) * 32;         // block size
    const int tid  = threadIdx.x;
    const int lane = tid & 31;
    const int wave = tid >> 5;
    const int row0 = blockIdx.x * 16;
    const int col0 = wave * 16;
    const int m    = lane & 15;               // A row in tile == C/D col in tile
    const int half = lane >> 4;
    const bool dual = (A2 != nullptr);

    // ---- stage 16x128 A tile(s) into LDS: 16B per lane, fully coalesced ----
    for (int i = tid; i < 16 * 32; i += NT) { // 512 float4 per tile
        const int r = i >> 5;
        const int c = (i & 31) * 4;
#if HAVE_ASYNC_LDS
        async_copy16(A + (size_t)(row0 + r) * 128 + c, &ldsA[0][r * LDS_STRIDE + c]);
        if (dual)
            async_copy16(A2 + (size_t)(row0 + r) * 128 + c, &ldsA[1][r * LDS_STRIDE + c]);
#else
        *(float4*)&ldsA[0][r * LDS_STRIDE + c] =
            *(const float4*)(A + (size_t)(row0 + r) * 128 + c);
        if (dual)
            *(float4*)&ldsA[1][r * LDS_STRIDE + c] =
                *(const float4*)(A2 + (size_t)(row0 + r) * 128 + c);
#endif
    }
#if HAVE_ASYNC_LDS
    asm volatile("s_wait_asynccnt 0x0" ::: "memory");
#endif
    __syncthreads();

    v8f c = {};
    {
        const float* a_lds = &ldsA[0][m * LDS_STRIDE];   // ds_load_b64 fragments
        const float* w_ptr = W + col0 + m;               // immediate-offset loads
#pragma unroll
        for (int k = 0; k < 128; k += 4) {
            const int ka = k + 2 * half;
            v2f a, b;
            a[0] = a_lds[ka];
            a[1] = a_lds[ka + 1];
            b[0] = w_ptr[ka * N];
            b[1] = w_ptr[(ka + 1) * N];
#if HAVE_WMMA_F32X4
            c = __builtin_amdgcn_wmma_f32_16x16x4_f32(false, a, false, b,
                                                      (short)0, c, false, false);
#else
            c[0] += a[0] * b[0] + a[1] * b[1];   // host-parse fallback only
#endif
        }
    }
    if (dual) {
        const float* a_lds = &ldsA[1][m * LDS_STRIDE];
        const float* w_ptr = W2 + col0 + m;
#pragma unroll
        for (int k = 0; k < 128; k += 4) {
            const int ka = k + 2 * half;
            v2f a, b;
            a[0] = a_lds[ka];
            a[1] = a_lds[ka + 1];
            b[0] = w_ptr[ka * N];
            b[1] = w_ptr[(ka + 1) * N];
#if HAVE_WMMA_F32X4
            c = __builtin_amdgcn_wmma_f32_16x16x4_f32(false, a, false, b,
                                                      (short)0, c, false, false);
#else
            c[0] += a[0] * b[0] + a[1] * b[1];
#endif
        }
    }

    // C/D layout: VGPR r, lane -> row = r + 8*half, col = lane&15.
    const int n  = col0 + m;
    const float bv = bias[n];
    float* orow = out + (size_t)(row0 + 8 * half) * N + n;
#pragma unroll
    for (int r = 0; r < 8; ++r) {
        float v = c[r] + bv;
        if (relu) v = fmaxf(v, 0.f);
        orow[(size_t)r * N] = v;
    }
}

// One wave per edge; lane handles 4 contiguous columns (float4 gather,
// 4x u32 atomicMax scatter). z >= 0 (relu), pooled pre-zeroed.
__global__ void edge_segmax(const int* __restrict__ src, const int* __restrict__ dst,
                            const float* __restrict__ z, unsigned int* __restrict__ pooled,
                            int E) {
    const int t = blockIdx.x * blockDim.x + threadIdx.x;
    const int e = t >> 5;
    if (e >= E) return;
    const int lane = t & 31;
    const int s = src[e];
    const int d = dst[e];
    const float4 v = *reinterpret_cast<const float4*>(z + (size_t)s * 128 + lane * 4);
    unsigned int* p = pooled + (size_t)d * 128 + lane * 4;
    atomicMax(p + 0, __float_as_uint(v.x));
    atomicMax(p + 1, __float_as_uint(v.y));
    atomicMax(p + 2, __float_as_uint(v.z));
    atomicMax(p + 3, __float_as_uint(v.w));
}

// o = maybe_relu(0.5 * (a + b))  (HeteroGraphConv 'mean' over the two dis dsts)
__global__ void combine_half(const float* __restrict__ a, const float* __restrict__ b,
                             float* __restrict__ o, int n, int relu) {
    const int i = blockIdx.x * blockDim.x + threadIdx.x;
    if (i >= n) return;
    float v = 0.5f * (a[i] + b[i]);
    if (relu) v = fmaxf(v, 0.f);
    o[i] = v;
}

static inline void launch_gemm(const float* A, const float* W, const float* A2,
                               const float* W2, const float* bias, float* out,
                               int M, int N, int relu, hipStream_t s) {
    if (N == 128)
        gemm128_wmma<128><<<M / 16, 256, 0, s>>>(A, W, A2, W2, bias, out, relu);
    else
        gemm128_wmma<64><<<M / 16, 128, 0, s>>>(A, W, A2, W2, bias, out, relu);
}

extern "C" void kernel_launch(void* const* d_in, const int* in_sizes, int n_in,
                              void* d_out, int out_size, void* d_ws, size_t ws_size,
                              hipStream_t stream) {
    const float* x_met = (const float*)d_in[0];
    const float* x_dis = (const float*)d_in[1];
    const int* src_mm  = (const int*)d_in[2];
    const int* dst_mm  = (const int*)d_in[3];
    const int* src_dd  = (const int*)d_in[4];
    const int* dst_dd  = (const int*)d_in[5];
    const int* src_md  = (const int*)d_in[6];
    const int* dst_md  = (const int*)d_in[7];
    const float* Wp1 = (const float*)d_in[8];
    const float* bp1 = (const float*)d_in[9];
    const float* Wn1 = (const float*)d_in[10];
    const float* Ws1 = (const float*)d_in[11];
    const float* b1  = (const float*)d_in[12];
    const float* Wp2 = (const float*)d_in[13];
    const float* bp2 = (const float*)d_in[14];
    const float* Wn2 = (const float*)d_in[15];
    const float* Ws2 = (const float*)d_in[16];
    const float* b2  = (const float*)d_in[17];

    const int NM  = in_sizes[0] / 128;   // 50000
    const int ND  = in_sizes[1] / 128;   // 50000
    const int E   = in_sizes[2];         // 800000
    const int OUT = 64;

    float* out = (float*)d_out;
    float* ws  = (float*)d_ws;
    const size_t SEG = (size_t)NM * 128; // 6.4M floats per buffer (NM == ND)
    float* z      = ws;
    float* pooled = ws + SEG;
    float* hmet   = ws + 2 * SEG;
    float* hdis   = ws + 3 * SEG;
    float* tA     = ws + 4 * SEG;
    float* tB     = ws + 5 * SEG;

    const int eblocks = (E * 32 + 255) / 256;
    auto pool = [&](const int* se, const int* de, int nDst) {
        (void)hipMemsetAsync(pooled, 0, (size_t)nDst * 128 * sizeof(float), stream);
        edge_segmax<<<eblocks, 256, 0, stream>>>(se, de, z, (unsigned int*)pooled, E);
    };

    // ----- Layer 1 (IN=128 -> HID=128); Wp/Wn/Ws stride 16384, b stride 128 --
    // rel 0: met->met
    launch_gemm(x_met, Wp1 + 0 * 16384, nullptr, nullptr, bp1 + 0 * 128, z, NM, 128, 1, stream);
    pool(src_mm, dst_mm, NM);
    launch_gemm(x_met, Ws1 + 0 * 16384, pooled, Wn1 + 0 * 16384, b1 + 0 * 128, hmet, NM, 128, 1, stream);
    // rel 1: dis->dis
    launch_gemm(x_dis, Wp1 + 1 * 16384, nullptr, nullptr, bp1 + 1 * 128, z, ND, 128, 1, stream);
    pool(src_dd, dst_dd, ND);
    launch_gemm(x_dis, Ws1 + 1 * 16384, pooled, Wn1 + 1 * 16384, b1 + 1 * 128, tA, ND, 128, 0, stream);
    // rel 2: met->dis
    launch_gemm(x_met, Wp1 + 2 * 16384, nullptr, nullptr, bp1 + 2 * 128, z, NM, 128, 1, stream);
    pool(src_md, dst_md, ND);
    launch_gemm(x_dis, Ws1 + 2 * 16384, pooled, Wn1 + 2 * 16384, b1 + 2 * 128, tB, ND, 128, 0, stream);
    combine_half<<<(ND * 128 + 255) / 256, 256, 0, stream>>>(tA, tB, hdis, ND * 128, 1);

    // ----- Layer 2 (HID=128 -> OUT=64); Wn2/Ws2 stride 8192, b2 stride 64 ---
    // rel 0: met->met -> h_met output
    launch_gemm(hmet, Wp2 + 0 * 16384, nullptr, nullptr, bp2 + 0 * 128, z, NM, 128, 1, stream);
    pool(src_mm, dst_mm, NM);
    launch_gemm(hmet, Ws2 + 0 * 8192, pooled, Wn2 + 0 * 8192, b2 + 0 * 64, out, NM, OUT, 0, stream);
    // rel 1: dis->dis
    launch_gemm(hdis, Wp2 + 1 * 16384, nullptr, nullptr, bp2 + 1 * 128, z, ND, 128, 1, stream);
    pool(src_dd, dst_dd, ND);
    launch_gemm(hdis, Ws2 + 1 * 8192, pooled, Wn2 + 1 * 8192, b2 + 1 * 64, tA, ND, OUT, 0, stream);
    // rel 2: met->dis
    launch_gemm(hmet, Wp2 + 2 * 16384, nullptr, nullptr, bp2 + 2 * 128, z, NM, 128, 1, stream);
    pool(src_md, dst_md, ND);
    launch_gemm(hdis, Ws2 + 2 * 8192, pooled, Wn2 + 2 * 8192, b2 + 2 * 64, tB, ND, OUT, 0, stream);
    combine_half<<<(ND * OUT + 255) / 256, 256, 0, stream>>>(tA, tB, out + (size_t)NM * OUT, ND * OUT, 0);
}